// ConvertParams0ToParamsGaussLayer_50002009260126
// MI455X (gfx1250) — compile-verified
//
#include <hip/hip_runtime.h>

// Problem constants (from reference): NV=64, NH=32, L=32
#define TNV  64
#define TNH  32
#define TL   32
#define N_HV 2048
#define N_HH 528
#define N_VV 2080

// LDS row strides (padded to dodge bank conflicts)
#define SLH 33   // 32x32 lower-tri chol_h2
#define SW  65   // 64-wide rows (and 32x64 / augmented 32x64)

typedef float v8f __attribute__((ext_vector_type(8)));
typedef float v4f __attribute__((ext_vector_type(4)));
typedef float v2f __attribute__((ext_vector_type(2)));

// row index i for flat lower-triangular index p (p = i*(i+1)/2 + j)
__device__ __forceinline__ int tri_row(int p) {
  int i = (int)((sqrtf(8.0f * (float)p + 1.0f) - 1.0f) * 0.5f);
  while ((((i + 1) * (i + 2)) >> 1) <= p) ++i;
  while (((i * (i + 1)) >> 1) > p) --i;
  return i;
}

// acc += sc[0:32].cs[0:32] + sc[32:64].cc[0:32], all via b128 loads.
// sc is LDS (16B aligned); cs/cc are 128B-aligned global rows.
__device__ __forceinline__ float dot_fourier(const float* __restrict__ sc,
                                             const float* __restrict__ cs,
                                             const float* __restrict__ cc,
                                             float acc) {
  const v4f* s4 = (const v4f*)sc;   // [0..7]=sin, [8..15]=cos
  const v4f* a4 = (const v4f*)cs;
  const v4f* b4 = (const v4f*)cc;
#pragma unroll
  for (int q = 0; q < 8; ++q) {
    v4f s = s4[q], a = a4[q];
    acc = fmaf(s.x, a.x, acc);
    acc = fmaf(s.y, a.y, acc);
    acc = fmaf(s.z, a.z, acc);
    acc = fmaf(s.w, a.w, acc);
  }
#pragma unroll
  for (int q = 0; q < 8; ++q) {
    v4f s = s4[8 + q], bb = b4[q];
    acc = fmaf(s.x, bb.x, acc);
    acc = fmaf(s.y, bb.y, acc);
    acc = fmaf(s.z, bb.z, acc);
    acc = fmaf(s.w, bb.w, acc);
  }
  return acc;
}

// 16x16 fp32 tile C += op(A) * op(B), K-accumulated with V_WMMA_F32_16X16X4_F32.
// A layout (ISA 7.12.2): lane m=lane&15, VGPR r holds K = r + 2*(lane>=16).
// B: lane n=lane&15, VGPR r holds K = r + 2*(lane>=16). C/D: VGPR v = row v+8*(lane>=16).
template <bool TA, bool TB>
__device__ __forceinline__ v8f wmma_acc(const float* A, int lda,
                                        const float* B, int ldb,
                                        int K, v8f c, int half, int l15) {
#pragma unroll 4
  for (int k0 = 0; k0 < K; k0 += 4) {
    const int ka = k0 + 2 * half;
    v2f a, b;
    a.x = TA ? A[(ka + 0) * lda + l15] : A[l15 * lda + (ka + 0)];
    a.y = TA ? A[(ka + 1) * lda + l15] : A[l15 * lda + (ka + 1)];
    b.x = TB ? B[l15 * ldb + (ka + 0)] : B[(ka + 0) * ldb + l15];
    b.y = TB ? B[l15 * ldb + (ka + 1)] : B[(ka + 1) * ldb + l15];
    c = __builtin_amdgcn_wmma_f32_16x16x4_f32(false, a, false, b, (short)0, c,
                                              false, false);
  }
  return c;
}

__global__ __launch_bounds__(256)
void gauss_convert_kernel(const float* __restrict__ tpt,
                          const float* __restrict__ mu_v1,
                          const float* __restrict__ chol_v_nz,
                          const float* __restrict__ freqs,
                          const float* __restrict__ muh_off,
                          const float* __restrict__ muh_sin,
                          const float* __restrict__ muh_cos,
                          const float* __restrict__ hv_off,
                          const float* __restrict__ hv_sin,
                          const float* __restrict__ hv_cos,
                          const float* __restrict__ hh_off,
                          const float* __restrict__ hh_sin,
                          const float* __restrict__ hh_cos,
                          float* __restrict__ out_mu,
                          float* __restrict__ out_chol) {
  extern __shared__ float sm[];
  float* s_sc = sm;                  // 64: [sin(32) | cos(32)]
  float* sLh  = s_sc + 64;           // 32 x SLH  (chol_h2, lower-tri)
  float* sLhv = sLh + 32 * SLH;      // 32 x SW   (chol_hv2, dense)
  float* sAug = sLhv + 32 * SW;      // 32 x SW   (GJ augmented [A|I])
  float* sM   = sAug + 32 * SW;      // 32 x SW   (A^-1 * Lhv)
  float* sA2  = sM + 32 * SW;        // 64 x SW   (amat2 -> cholesky L in place)
  float* sInv = sA2 + 64 * SW;       // 64 x SW   (inv of L)
  float* sLv1 = sAug;                // reuse sAug+sM (2*32*SW == 64*SW) late
  float* sStage = sInv;              // borrow sInv region early for async copy

  const int b    = blockIdx.x;
  const int tid  = threadIdx.x;
  const int w    = tid >> 5;
  const int lane = tid & 31;
  const int half = lane >> 4;
  const int l15  = lane & 15;

  // ---- kick off async global->LDS copy of this batch's chol_v_nz (8.3KB).
  // Latency hides behind the Fourier + Gauss-Jordan phases below.
  {
    const float* src = chol_v_nz + (size_t)b * N_VV;
    unsigned ldsbase = (unsigned)(uintptr_t)sStage;
    for (int p = tid; p < N_VV; p += 256) {
      unsigned laddr = ldsbase + 4u * (unsigned)p;
      const float* g = src + p;
      asm volatile("global_load_async_to_lds_b32 %0, %1, off"
                   :
                   : "v"(laddr), "v"(g)
                   : "memory");
    }
  }

  // ---- phases ----
  if (tid < TL) {
    float ph = tpt[b] * freqs[tid];
    s_sc[tid]      = sinf(ph);
    s_sc[tid + TL] = cosf(ph);
  }
  for (int i = tid; i < 32 * SLH; i += 256) sLh[i] = 0.0f;
  __syncthreads();

  // ---- Fourier latents ----
  for (int p = tid; p < N_HV; p += 256) {          // hv: dense 32x64
    float acc = dot_fourier(s_sc, hv_sin + p * TL, hv_cos + p * TL, hv_off[p]);
    sLhv[(p >> 6) * SW + (p & 63)] = acc;
  }
  for (int p = tid; p < N_HH; p += 256) {          // hh: lower-tri 32x32
    float acc = dot_fourier(s_sc, hh_sin + p * TL, hh_cos + p * TL, hh_off[p]);
    int i = tri_row(p);
    sLh[i * SLH + (p - ((i * (i + 1)) >> 1))] = acc;
  }
  // ---- mu outputs ----
  float* omu = out_mu + (size_t)b * 96;
  if (tid < TNV) omu[tid] = mu_v1[(size_t)b * TNV + tid];
  if (tid >= 64 && tid < 96) {
    int h = tid - 64;
    omu[64 + h] =
        dot_fourier(s_sc, muh_sin + h * TL, muh_cos + h * TL, muh_off[h]);
  }
  __syncthreads();

  // ---- amat = Lh*Lh^T + Lhv*Lhv^T -> sAug[:,0:32]; I -> sAug[:,32:64] ----
  if (w < 4) {
    int ti = w >> 1, tj = w & 1;
    v8f c = {};
    c = wmma_acc<false, true>(sLh + ti * 16 * SLH, SLH,
                              sLh + tj * 16 * SLH, SLH, TNH, c, half, l15);
    c = wmma_acc<false, true>(sLhv + ti * 16 * SW, SW,
                              sLhv + tj * 16 * SW, SW, TNV, c, half, l15);
#pragma unroll
    for (int v = 0; v < 8; ++v)
      sAug[(ti * 16 + v + 8 * half) * SW + tj * 16 + l15] = c[v];
  } else {
    int q = tid - 128;
    for (int e = q; e < 32 * 32; e += 128) {
      int i = e >> 5, j = e & 31;
      sAug[i * SW + 32 + j] = (i == j) ? 1.0f : 0.0f;
    }
  }
  __syncthreads();

  // ---- Gauss-Jordan: sAug[:,32:64] <- amat^-1 ----
  {
    const int gi = tid >> 3;          // row 0..31
    const int gc = (tid & 7) << 3;    // 8-col strip
    for (int k = 0; k < 32; ++k) {
      float pinv = 1.0f / sAug[k * SW + k];
      __syncthreads();
      if (tid < 64) sAug[k * SW + tid] *= pinv;
      __syncthreads();
      float fac = sAug[gi * SW + k];
      __syncthreads();
      if (gi != k) {
#pragma unroll
        for (int j = 0; j < 8; ++j)
          sAug[gi * SW + gc + j] =
              fmaf(-fac, sAug[k * SW + gc + j], sAug[gi * SW + gc + j]);
      }
      __syncthreads();
    }
  }

  // ---- M = amat^-1 @ Lhv (32x64), 8 tiles across 8 waves ----
  {
    int ti = w >> 2, tj = w & 3;
    v8f c = {};
    c = wmma_acc<false, false>(sAug + ti * 16 * SW + 32, SW,
                               sLhv + tj * 16, SW, TNH, c, half, l15);
#pragma unroll
    for (int v = 0; v < 8; ++v)
      sM[(ti * 16 + v + 8 * half) * SW + tj * 16 + l15] = c[v];
  }
  __syncthreads();

  // ---- amat2 = I - Lhv^T @ M (64x64), 16 tiles ----
#pragma unroll
  for (int tt = 0; tt < 2; ++tt) {
    int t = w + 8 * tt;
    int ti = t >> 2, tj = t & 3;
    v8f c = {};
    c = wmma_acc<true, false>(sLhv + ti * 16, SW, sM + tj * 16, SW, TNH, c,
                              half, l15);
#pragma unroll
    for (int v = 0; v < 8; ++v) {
      int gr = ti * 16 + v + 8 * half;
      int gn = tj * 16 + l15;
      sA2[gr * SW + gn] = ((gr == gn) ? 1.0f : 0.0f) - c[v];
    }
  }
  // Drain this wave's async copy; barrier below publishes all waves' LDS data.
  asm volatile("s_wait_asynccnt 0x0" ::: "memory");
  __syncthreads();

  // ---- scatter chol_v1 (lower-tri 64x64) from staged LDS into sLv1 ----
  for (int i = tid; i < 64 * SW; i += 256) sLv1[i] = 0.0f;
  __syncthreads();
  for (int p = tid; p < N_VV; p += 256) {
    int i = tri_row(p);
    sLv1[i * SW + (p - ((i * (i + 1)) >> 1))] = sStage[p];
  }
  __syncthreads();

  // ---- Cholesky of sA2 (in place, lower) ----
  for (int k = 0; k < 64; ++k) {
    if (tid == 0) sA2[k * SW + k] = sqrtf(sA2[k * SW + k]);
    __syncthreads();
    float dinv = 1.0f / sA2[k * SW + k];
    if (tid > k && tid < 64) sA2[tid * SW + k] *= dinv;
    __syncthreads();
    int n = 63 - k;
    int total = (n * (n + 1)) >> 1;
    for (int q = tid; q < total; q += 256) {
      int a = tri_row(q);
      int j = q - ((a * (a + 1)) >> 1);
      int gi2 = k + 1 + a, gj2 = k + 1 + j;
      sA2[gi2 * SW + gj2] = fmaf(-sA2[gi2 * SW + k], sA2[gj2 * SW + k],
                                 sA2[gi2 * SW + gj2]);
    }
    __syncthreads();
  }

  // ---- sInv = inv(L) : column-parallel forward substitution ----
  // (overwrites the staging region; staged data fully consumed above)
  if (tid < 64) {
    int j = tid;
    for (int i = 0; i < j; ++i) sInv[i * SW + j] = 0.0f;
    sInv[j * SW + j] = 1.0f / sA2[j * SW + j];
    for (int i = j + 1; i < 64; ++i) {
      float s = 0.0f;
      for (int k = j; k < i; ++k)
        s = fmaf(sA2[i * SW + k], sInv[k * SW + j], s);
      sInv[i * SW + j] = -s / sA2[i * SW + i];
    }
  }
  __syncthreads();

  // ---- chol_v2 = chol_v1 @ inv(L) -> global rows 0..63, cols 0..63 ----
  float* och = out_chol + (size_t)b * 96 * 96;
#pragma unroll
  for (int tt = 0; tt < 2; ++tt) {
    int t = w + 8 * tt;
    int ti = t >> 2, tj = t & 3;
    v8f c = {};
    c = wmma_acc<false, false>(sLv1 + ti * 16 * SW, SW, sInv + tj * 16, SW,
                               TNV, c, half, l15);
#pragma unroll
    for (int v = 0; v < 8; ++v)
      och[(ti * 16 + v + 8 * half) * 96 + tj * 16 + l15] = c[v];
  }

  // ---- remaining output blocks ----
  for (int e = tid; e < 64 * 32; e += 256) {        // rows 0..63, cols 64..95
    int r = e >> 5, cc2 = e & 31;
    och[r * 96 + 64 + cc2] = 0.0f;
  }
  for (int e = tid; e < 32 * 96; e += 256) {        // rows 64..95
    int r = e / 96, cc2 = e - r * 96;
    float val = (cc2 < 64) ? sLhv[r * SW + cc2] : sLh[r * SLH + (cc2 - 64)];
    och[(64 + r) * 96 + cc2] = val;
  }
}

extern "C" void kernel_launch(void* const* d_in, const int* in_sizes, int n_in,
                              void* d_out, int out_size, void* d_ws,
                              size_t ws_size, hipStream_t stream) {
  const float* tpt   = (const float*)d_in[0];
  const float* mu_v1 = (const float*)d_in[1];
  const float* cvnz  = (const float*)d_in[2];
  const float* freqs = (const float*)d_in[3];
  const float* muh_o = (const float*)d_in[4];
  const float* muh_s = (const float*)d_in[5];
  const float* muh_c = (const float*)d_in[6];
  const float* hv_o  = (const float*)d_in[7];
  const float* hv_s  = (const float*)d_in[8];
  const float* hv_c  = (const float*)d_in[9];
  const float* hh_o  = (const float*)d_in[10];
  const float* hh_s  = (const float*)d_in[11];
  const float* hh_c  = (const float*)d_in[12];
  const int B = in_sizes[0];

  float* out_mu   = (float*)d_out;
  float* out_chol = out_mu + (size_t)B * 96;

  // 64 + 32*33 + 3*(32*65) + 2*(64*65) floats = 15680 floats = 62720 B (<64KB)
  const size_t shmem =
      (size_t)(64 + 32 * SLH + 3 * 32 * SW + 2 * 64 * SW) * sizeof(float);

  gauss_convert_kernel<<<B, 256, shmem, stream>>>(
      tpt, mu_v1, cvnz, freqs, muh_o, muh_s, muh_c, hv_o, hv_s, hv_c, hh_o,
      hh_s, hh_c, out_mu, out_chol);
}